// RGCN_DDI_Model_9517647528309
// MI455X (gfx1250) — compile-verified
//
#include <hip/hip_runtime.h>

typedef __attribute__((ext_vector_type(2))) float v2f;
typedef __attribute__((ext_vector_type(8))) float v8f;

#define NUM_NODES 50000
#define NUM_REL   32
#define EMB       16
#define HID       64
#define NUM_EDGES 1600000
#define NUM_PAIRS 16384

#define NPAD    50048              // nodes padded to multiple of 32
#define BLOCKN  12512              // dst-node block for layer-2 aggregation (4 blocks)
#define CNT_F   ((size_t)NPAD * NUM_REL)             // 1,601,536 floats
#define SUMS_F  ((size_t)BLOCKN * NUM_REL * HID)     // 25,624,576 floats (>= NPAD*R*EMB)
#define X1_F    ((size_t)NPAD * HID)
#define X2_F    ((size_t)NUM_NODES * HID)

static __device__ __forceinline__ v8f wmma4(v2f a, v2f b, v8f c) {
  // D = A(16x4,f32) x B(4x16,f32) + C(16x16,f32)
  return __builtin_amdgcn_wmma_f32_16x16x4_f32(false, a, false, b, (short)0, c,
                                               false, false);
}

__global__ void zero_f32(float* __restrict__ p, long long n) {
  long long i = (long long)blockIdx.x * blockDim.x + threadIdx.x;
  if (i < n) p[i] = 0.0f;
}

__global__ void count_edges(const int* __restrict__ ei, const int* __restrict__ et,
                            float* __restrict__ cnt) {
  int e = blockIdx.x * blockDim.x + threadIdx.x;
  if (e >= NUM_EDGES) return;
  int dst = ei[NUM_EDGES + e];
  int r   = et[e];
  atomicAdd(&cnt[(size_t)dst * NUM_REL + r], 1.0f);
}

// cnt[i] -> 1 / max(cnt[i], 1)   (exact division once; hot loops just multiply)
__global__ void invert_cnt(float* __restrict__ c, long long n) {
  long long i = (long long)blockIdx.x * blockDim.x + threadIdx.x;
  if (i < n) c[i] = 1.0f / fmaxf(c[i], 1.0f);
}

// Layer-1 segment-sum of raw 16-dim embeddings into [N*R, 16]
__global__ void scatter1(const int* __restrict__ ei, const int* __restrict__ et,
                         const float* __restrict__ x, float* __restrict__ sums) {
  int e = blockIdx.x * blockDim.x + threadIdx.x;
  if (e >= NUM_EDGES) return;
  int src = ei[e];
  int dst = ei[NUM_EDGES + e];
  int r   = et[e];
  float* s        = sums + ((size_t)dst * NUM_REL + r) * EMB;
  const float* xs = x + (size_t)src * EMB;
#pragma unroll
  for (int j = 0; j < EMB; j++) atomicAdd(&s[j], xs[j]);
}

// Layer-2 segment-sum (dst-node blocked) of 64-dim features; 4 threads/edge
__global__ void scatter2(const int* __restrict__ ei, const int* __restrict__ et,
                         const float* __restrict__ x1, float* __restrict__ sums, int lo) {
  int t = blockIdx.x * blockDim.x + threadIdx.x;
  int e = t >> 2, part = t & 3;
  if (e >= NUM_EDGES) return;
  int dst = ei[NUM_EDGES + e];
  if (dst < lo || dst >= lo + BLOCKN) return;
  int src = ei[e];
  int r   = et[e];
  float* s        = sums + ((size_t)(dst - lo) * NUM_REL + r) * HID + part * 16;
  const float* xs = x1 + (size_t)src * HID + part * 16;
#pragma unroll
  for (int j = 0; j < 16; j++) atomicAdd(&s[j], xs[j]);
}

// x1 = relu( sum_r mean[n,r,:16] @ W1_r + x0 @ root1 + b1 )
// one wave per 32 nodes (two 16-row M-tiles share every B load)
__global__ __launch_bounds__(256) void gemm1(const float* __restrict__ sums,
                                             const float* __restrict__ icnt,
                                             const float* __restrict__ x0,
                                             const float* __restrict__ W1,
                                             const float* __restrict__ root1,
                                             const float* __restrict__ b1,
                                             float* __restrict__ x1) {
  int wave = threadIdx.x >> 5, lane = threadIdx.x & 31;
  int pt = blockIdx.x * (blockDim.x >> 5) + wave;
  if (pt >= NPAD / 32) return;              // 1564 tile-pairs, uniform per wave
  int m = lane & 15, half = lane >> 4;
  int row0 = pt * 32 + m;                   // < NPAD
  int row1 = row0 + 16;
  v8f acc0[4] = {}, acc1[4] = {};
  for (int r = 0; r < NUM_REL; r++) {
    float s0 = icnt[(size_t)row0 * NUM_REL + r];
    float s1 = icnt[(size_t)row1 * NUM_REL + r];
    const float* A0 = sums + ((size_t)row0 * NUM_REL + r) * EMB;
    const float* A1 = sums + ((size_t)row1 * NUM_REL + r) * EMB;
#pragma unroll
    for (int k = 0; k < 4; k++) {
      int col = k * 4 + half * 2;
      v2f a0; a0.x = A0[col] * s0; a0.y = A0[col + 1] * s0;
      v2f a1; a1.x = A1[col] * s1; a1.y = A1[col + 1] * s1;
#pragma unroll
      for (int t = 0; t < 4; t++) {
        int n = t * 16 + m;
        v2f b;
        b.x = W1[((size_t)r * EMB + col) * HID + n];
        b.y = W1[((size_t)r * EMB + col + 1) * HID + n];
        acc0[t] = wmma4(a0, b, acc0[t]);
        acc1[t] = wmma4(a1, b, acc1[t]);
      }
    }
  }
  // root path (clamp pad rows; their x1 output is well-defined but unused downstream)
  int rc0 = row0 < NUM_NODES ? row0 : NUM_NODES - 1;
  int rc1 = row1 < NUM_NODES ? row1 : NUM_NODES - 1;
  const float* X0 = x0 + (size_t)rc0 * EMB;
  const float* X1r = x0 + (size_t)rc1 * EMB;
#pragma unroll
  for (int k = 0; k < 4; k++) {
    int col = k * 4 + half * 2;
    v2f a0; a0.x = X0[col]; a0.y = X0[col + 1];
    v2f a1; a1.x = X1r[col]; a1.y = X1r[col + 1];
#pragma unroll
    for (int t = 0; t < 4; t++) {
      int n = t * 16 + m;
      v2f b; b.x = root1[(size_t)col * HID + n]; b.y = root1[(size_t)(col + 1) * HID + n];
      acc0[t] = wmma4(a0, b, acc0[t]);
      acc1[t] = wmma4(a1, b, acc1[t]);
    }
  }
#pragma unroll
  for (int t = 0; t < 4; t++) {
    int n = t * 16 + m;
    float bias = b1[n];
#pragma unroll
    for (int v = 0; v < 8; v++) {
      int rr = pt * 32 + half * 8 + v;
      x1[(size_t)rr * HID + n]        = fmaxf(acc0[t][v] + bias, 0.0f);
      x1[(size_t)(rr + 16) * HID + n] = fmaxf(acc1[t][v] + bias, 0.0f);
    }
  }
}

// x2 = sum_r mean[n,r,:64] @ W2_r + x1 @ root2 + b2  (dst-node blocked, 2 M-tiles/wave)
__global__ __launch_bounds__(256) void gemm2(const float* __restrict__ sums,
                                             const float* __restrict__ icnt,
                                             const float* __restrict__ x1,
                                             const float* __restrict__ W2,
                                             const float* __restrict__ root2,
                                             const float* __restrict__ b2,
                                             float* __restrict__ x2, int lo) {
  int wave = threadIdx.x >> 5, lane = threadIdx.x & 31;
  int pt = blockIdx.x * (blockDim.x >> 5) + wave;
  if (pt >= BLOCKN / 32) return;            // 391 tile-pairs
  int m = lane & 15, half = lane >> 4;
  int lr0 = pt * 32 + m;                    // local rows < BLOCKN
  int lr1 = lr0 + 16;
  int row0 = lo + lr0, row1 = lo + lr1;     // < NPAD always
  v8f acc0[4] = {}, acc1[4] = {};
  for (int r = 0; r < NUM_REL; r++) {
    float s0 = icnt[(size_t)row0 * NUM_REL + r];
    float s1 = icnt[(size_t)row1 * NUM_REL + r];
    const float* A0 = sums + ((size_t)lr0 * NUM_REL + r) * HID;
    const float* A1 = sums + ((size_t)lr1 * NUM_REL + r) * HID;
#pragma unroll
    for (int k = 0; k < 16; k++) {
      int col = k * 4 + half * 2;
      v2f a0; a0.x = A0[col] * s0; a0.y = A0[col + 1] * s0;
      v2f a1; a1.x = A1[col] * s1; a1.y = A1[col + 1] * s1;
#pragma unroll
      for (int t = 0; t < 4; t++) {
        int n = t * 16 + m;
        v2f b;
        b.x = W2[((size_t)r * HID + col) * HID + n];
        b.y = W2[((size_t)r * HID + col + 1) * HID + n];
        acc0[t] = wmma4(a0, b, acc0[t]);
        acc1[t] = wmma4(a1, b, acc1[t]);
      }
    }
  }
  const float* X0 = x1 + (size_t)row0 * HID;  // x1 has NPAD well-defined rows
  const float* X1r = x1 + (size_t)row1 * HID;
#pragma unroll
  for (int k = 0; k < 16; k++) {
    int col = k * 4 + half * 2;
    v2f a0; a0.x = X0[col]; a0.y = X0[col + 1];
    v2f a1; a1.x = X1r[col]; a1.y = X1r[col + 1];
#pragma unroll
    for (int t = 0; t < 4; t++) {
      int n = t * 16 + m;
      v2f b; b.x = root2[(size_t)col * HID + n]; b.y = root2[(size_t)(col + 1) * HID + n];
      acc0[t] = wmma4(a0, b, acc0[t]);
      acc1[t] = wmma4(a1, b, acc1[t]);
    }
  }
#pragma unroll
  for (int t = 0; t < 4; t++) {
    int n = t * 16 + m;
    float bias = b2[n];
#pragma unroll
    for (int v = 0; v < 8; v++) {
      int g0 = lo + pt * 32 + half * 8 + v;
      int g1 = g0 + 16;
      if (g0 < NUM_NODES) x2[(size_t)g0 * HID + n] = acc0[t][v] + bias;
      if (g1 < NUM_NODES) x2[(size_t)g1 * HID + n] = acc1[t][v] + bias;
    }
  }
}

// out = relu(concat(x2[d1],x2[d2]) @ cls_w1 + cls_b1) @ cls_w2 + cls_b2
__global__ __launch_bounds__(256) void classifier(const float* __restrict__ x2,
                                                  const int* __restrict__ d1,
                                                  const int* __restrict__ d2,
                                                  const float* __restrict__ w1,
                                                  const float* __restrict__ cb1,
                                                  const float* __restrict__ w2,
                                                  const float* __restrict__ cb2,
                                                  float* __restrict__ out) {
  __shared__ float lds[8][16 * 66];
  int wave = threadIdx.x >> 5, lane = threadIdx.x & 31;
  int tile = blockIdx.x * (blockDim.x >> 5) + wave;   // exactly 1024 tiles / 128 blocks
  int m = lane & 15, half = lane >> 4;
  int pair = tile * 16 + m;
  int i1 = d1[pair], i2 = d2[pair];
  v8f acc[4] = {};
#pragma unroll
  for (int k = 0; k < 32; k++) {            // K = 128 (concat)
    int col = k * 4 + half * 2;
    const float* X = (k < 16) ? (x2 + (size_t)i1 * HID) : (x2 + (size_t)i2 * HID);
    int cc = (k < 16) ? col : (col - 64);
    v2f a; a.x = X[cc]; a.y = X[cc + 1];
#pragma unroll
    for (int t = 0; t < 4; t++) {
      int n = t * 16 + m;
      v2f b; b.x = w1[(size_t)col * HID + n]; b.y = w1[(size_t)(col + 1) * HID + n];
      acc[t] = wmma4(a, b, acc[t]);
    }
  }
  float* L = lds[wave];
#pragma unroll
  for (int t = 0; t < 4; t++) {
    int n = t * 16 + m;
    float bias = cb1[n];
#pragma unroll
    for (int v = 0; v < 8; v++)
      L[(half * 8 + v) * 66 + n] = fmaxf(acc[t][v] + bias, 0.0f);
  }
  __syncthreads();                           // all waves active; safe
  v8f acc2[2] = {};
#pragma unroll
  for (int k = 0; k < 16; k++) {             // K = 64
    int col = k * 4 + half * 2;
    v2f a; a.x = L[m * 66 + col]; a.y = L[m * 66 + col + 1];
#pragma unroll
    for (int t = 0; t < 2; t++) {
      int n = t * 16 + m;
      v2f b; b.x = w2[(size_t)col * NUM_REL + n]; b.y = w2[(size_t)(col + 1) * NUM_REL + n];
      acc2[t] = wmma4(a, b, acc2[t]);
    }
  }
#pragma unroll
  for (int t = 0; t < 2; t++) {
    int n = t * 16 + m;
    float bias = cb2[n];
#pragma unroll
    for (int v = 0; v < 8; v++)
      out[(size_t)(tile * 16 + half * 8 + v) * NUM_REL + n] = acc2[t][v] + bias;
  }
}

extern "C" void kernel_launch(void* const* d_in, const int* in_sizes, int n_in,
                              void* d_out, int out_size, void* d_ws, size_t ws_size,
                              hipStream_t stream) {
  const int*   edge_index = (const int*)d_in[0];
  const int*   edge_type  = (const int*)d_in[1];
  const int*   drug1_idx  = (const int*)d_in[2];
  const int*   drug2_idx  = (const int*)d_in[3];
  const float* node_emb   = (const float*)d_in[4];
  const float* W1    = (const float*)d_in[5];
  const float* root1 = (const float*)d_in[6];
  const float* b1    = (const float*)d_in[7];
  const float* W2    = (const float*)d_in[8];
  const float* root2 = (const float*)d_in[9];
  const float* b2    = (const float*)d_in[10];
  const float* cw1   = (const float*)d_in[11];
  const float* cb1   = (const float*)d_in[12];
  const float* cw2   = (const float*)d_in[13];
  const float* cb2   = (const float*)d_in[14];
  float* out = (float*)d_out;

  float* ws   = (float*)d_ws;
  float* cnt  = ws;                 // CNT_F (holds inverse counts after invert_cnt)
  float* sums = cnt + CNT_F;        // SUMS_F (shared by layer-1 and layer-2 blocks)
  float* x1   = sums + SUMS_F;      // X1_F
  float* x2   = x1 + X1_F;          // X2_F

  const int TB = 256;
  auto blocks = [](long long n, int tb) { return (unsigned)((n + tb - 1) / tb); };

  // counts -> inverse counts
  zero_f32<<<blocks((long long)CNT_F, TB), TB, 0, stream>>>(cnt, (long long)CNT_F);
  count_edges<<<blocks(NUM_EDGES, TB), TB, 0, stream>>>(edge_index, edge_type, cnt);
  invert_cnt<<<blocks((long long)CNT_F, TB), TB, 0, stream>>>(cnt, (long long)CNT_F);

  // layer 1: segment-sum -> WMMA GEMM
  zero_f32<<<blocks((long long)SUMS_F, TB), TB, 0, stream>>>(sums, (long long)SUMS_F);
  scatter1<<<blocks(NUM_EDGES, TB), TB, 0, stream>>>(edge_index, edge_type, node_emb, sums);
  gemm1<<<blocks(NPAD / 32, 8), TB, 0, stream>>>(sums, cnt, node_emb, W1, root1, b1, x1);

  // layer 2: blocked over dst nodes
  for (int b = 0; b < 4; b++) {
    int lo = b * BLOCKN;
    zero_f32<<<blocks((long long)SUMS_F, TB), TB, 0, stream>>>(sums, (long long)SUMS_F);
    scatter2<<<blocks((long long)NUM_EDGES * 4, TB), TB, 0, stream>>>(edge_index, edge_type,
                                                                      x1, sums, lo);
    gemm2<<<blocks(BLOCKN / 32, 8), TB, 0, stream>>>(sums, cnt, x1, W2, root2, b2, x2, lo);
  }

  // classifier MLP
  classifier<<<blocks(NUM_PAIRS / 16, 8), TB, 0, stream>>>(x2, drug1_idx, drug2_idx,
                                                           cw1, cb1, cw2, cb2, out);
}